// GAT_89507118448839
// MI455X (gfx1250) — compile-verified
//
#include <hip/hip_runtime.h>
#include <hip/hip_bf16.h>

// ---------------------------------------------------------------------------
// 2-layer GAT for MI455X (gfx1250, wave32).
// Memory/atomic-bound (~8 GFLOP dense vs ~1 GB traffic) -> fp32 throughout,
// GEMMs via V_WMMA_F32_16X16X4_F32, A tiles staged to LDS by the Tensor Data
// Mover (with hardware row padding for bank-conflict-free reads), segment
// softmax / scatter-aggregate via L2-resident f32 atomics.
// ---------------------------------------------------------------------------

typedef __attribute__((ext_vector_type(2))) float v2f;
typedef __attribute__((ext_vector_type(8))) float v8f;
typedef __attribute__((ext_vector_type(4))) unsigned int u32x4;
typedef __attribute__((ext_vector_type(8))) int i32x8;
typedef __attribute__((ext_vector_type(4))) int i32x4;

#define GAT_IN  256
#define GAT_HID 128     // H0*D0 = 4*32
#define GAT_H0  4
#define GAT_D0  32
#define GAT_C   40
#define SLOPE   0.2f

// monotone float -> int map (involution), enables atomicMax on ints
__device__ __forceinline__ int f2oi(float f) {
    int i = __float_as_int(f);
    return i >= 0 ? i : (i ^ 0x7FFFFFFF);
}
__device__ __forceinline__ float oi2f(int i) {
    return __int_as_float(i >= 0 ? i : (i ^ 0x7FFFFFFF));
}

__global__ void fill_i32_kernel(int* p, int v, int count) {
    int i = blockIdx.x * blockDim.x + threadIdx.x;
    if (i < count) p[i] = v;
}

// ---------------------------------------------------------------------------
// TDM: DMA a 2D tile of 16 rows x 256 f32 from global into LDS, padding the
// LDS destination by 1 DWORD every 256 DWORDs (row stride 257 floats -> all
// 16 rows land in distinct banks). Rows beyond `tensor_rows` read as zero.
// D# layout per CDNA5 ISA ch. 8 (group0: count/lds/global addr; group1:
// data_size=4B, pad_enable, pad_interval=256dw, pad_amount=1dw, dims).
// This toolchain exposes the 6-arg builtin (g0, g1, g2, g3, g_extra, cpol).
// ---------------------------------------------------------------------------
__device__ __forceinline__ void tdm_load_tile_16x256(
    unsigned lds_byte_off, const float* gsrc, unsigned tensor_rows)
{
    unsigned long long ga = (unsigned long long)(const void*)gsrc;
    u32x4 g0;
    g0[0] = 1u;                                         // count=1, is_restore=0
    g0[1] = lds_byte_off;                               // lds_addr
    g0[2] = (unsigned)(ga & 0xFFFFFFFFu);               // global_addr[31:0]
    g0[3] = (unsigned)((ga >> 32) & 0x1FFFFFFu)         // global_addr[56:32]
          | (2u << 30);                                 // type=2 ("image")
    i32x8 g1;
    g1[0] = (2 << 16) | (1 << 20) | (7 << 22);          // data_size=4B, pad_en, interval=256dw, amount=1dw
    g1[1] = (int)(256u << 16);                          // tensor_dim0 lo16 (=256); barrier addr 0
    g1[2] = (int)((tensor_rows & 0xFFFFu) << 16);       // tensor_dim0 hi16=0 | tensor_dim1 lo16
    g1[3] = (int)(((tensor_rows >> 16) & 0xFFFFu)       // tensor_dim1 hi16
          | (256u << 16));                              // tile_dim0 = 256
    g1[4] = 16;                                         // tile_dim1=16, tile_dim2=0
    g1[5] = 256;                                        // tensor_dim0_stride lo32
    g1[6] = 0;                                          // stride hi | dim1_stride lo
    g1[7] = 0;
    i32x4 z4 = {};
    i32x8 z8 = {};
    __builtin_amdgcn_tensor_load_to_lds(g0, g1, z4, z4, z8, 0);
}

// ---------------------------------------------------------------------------
// GEMM0: feat0[n,128] = x[n,256] @ W0[256,128]
// Block = 256 threads (8 waves). Block owns 16 rows; wave w owns 16-col tile.
// A tile DMA'd to LDS by the TDM (row stride 257 via hardware padding).
// ---------------------------------------------------------------------------
__global__ __launch_bounds__(256) void gemm0_kernel(
    const float* __restrict__ x, const float* __restrict__ W,
    float* __restrict__ feat, int n)
{
    __shared__ float As[16 * 257];
    const int t    = threadIdx.x;
    const int row0 = blockIdx.x * 16;

    if (t == 0) {
        tdm_load_tile_16x256((unsigned)(unsigned long long)(void*)As,
                             x + (size_t)row0 * GAT_IN,
                             (unsigned)(n - row0));
        __builtin_amdgcn_s_wait_tensorcnt(0);
    }
    __syncthreads();
    asm volatile("" : : "r"(As) : "memory");  // LDS written by TDM, not by threads

    const int wave = t >> 5;
    const int lane = t & 31;
    const int half = lane >> 4;     // 0: K pair {0,1}, 1: K pair {2,3}
    const int l16  = lane & 15;
    const int col  = wave * 16 + l16;

    v8f acc = {};
    for (int k0 = 0; k0 < GAT_IN; k0 += 4) {
        int k = k0 + half * 2;
        v2f a, b;
        a.x = As[l16 * 257 + k];
        a.y = As[l16 * 257 + k + 1];
        b.x = W[k * GAT_HID + col];
        b.y = W[(k + 1) * GAT_HID + col];
        acc = __builtin_amdgcn_wmma_f32_16x16x4_f32(
            false, a, false, b, (short)0, acc, false, false);
    }
    if (row0 + 16 <= n) {           // full tile: branch-free store run
        #pragma unroll
        for (int v = 0; v < 8; ++v)
            feat[(size_t)(row0 + half * 8 + v) * GAT_HID + col] = acc[v];
    } else {
        #pragma unroll
        for (int v = 0; v < 8; ++v) {
            int gr = row0 + half * 8 + v;
            if (gr < n) feat[(size_t)gr * GAT_HID + col] = acc[v];
        }
    }
}

// ---------------------------------------------------------------------------
// GEMM1: feat1[n,40] = h[n,128] @ W1[128,40]  (cols padded to 48, guarded)
// ---------------------------------------------------------------------------
__global__ __launch_bounds__(256) void gemm1_kernel(
    const float* __restrict__ h, const float* __restrict__ W,
    float* __restrict__ feat, int n)
{
    __shared__ float As[16 * 129];
    const int t    = threadIdx.x;
    const int row0 = blockIdx.x * 16;

    #pragma unroll
    for (int i = 0; i < 8; ++i) {           // 16x128 = 2048 elements
        int idx = i * 256 + t;
        int r = idx >> 7, c = idx & 127;
        int gr = row0 + r;
        As[r * 129 + c] = (gr < n) ? h[(size_t)gr * GAT_HID + c] : 0.f;
    }
    __syncthreads();

    const int wave = t >> 5;
    if (wave >= 3) return;                  // 3 col tiles cover 48 >= 40
    const int lane = t & 31;
    const int half = lane >> 4;
    const int l16  = lane & 15;
    const int col  = wave * 16 + l16;
    const bool cok = (col < GAT_C);

    v8f acc = {};
    for (int k0 = 0; k0 < GAT_HID; k0 += 4) {
        int k = k0 + half * 2;
        v2f a, b;
        a.x = As[l16 * 129 + k];
        a.y = As[l16 * 129 + k + 1];
        b.x = cok ? W[k * GAT_C + col] : 0.f;
        b.y = cok ? W[(k + 1) * GAT_C + col] : 0.f;
        acc = __builtin_amdgcn_wmma_f32_16x16x4_f32(
            false, a, false, b, (short)0, acc, false, false);
    }
    if (cok) {
        if (row0 + 16 <= n) {
            #pragma unroll
            for (int v = 0; v < 8; ++v)
                feat[(size_t)(row0 + half * 8 + v) * GAT_C + col] = acc[v];
        } else {
            #pragma unroll
            for (int v = 0; v < 8; ++v) {
                int gr = row0 + half * 8 + v;
                if (gr < n) feat[(size_t)gr * GAT_C + col] = acc[v];
            }
        }
    }
}

// per-(node,head) attention logits for layer 0
__global__ void lr0_kernel(const float* __restrict__ feat,
                           const float* __restrict__ al, const float* __restrict__ ar,
                           float* __restrict__ el, float* __restrict__ er, int n)
{
    int tid = blockIdx.x * blockDim.x + threadIdx.x;
    if (tid >= n * GAT_H0) return;
    int nid = tid >> 2, hh = tid & 3;
    const float* f = feat + (size_t)nid * GAT_HID + hh * GAT_D0;
    const float* a = al + hh * GAT_D0;
    const float* b = ar + hh * GAT_D0;
    float sl = 0.f, sr = 0.f;
    #pragma unroll
    for (int d = 0; d < GAT_D0; ++d) { float v = f[d]; sl += v * a[d]; sr += v * b[d]; }
    el[tid] = sl; er[tid] = sr;
}

// per-node logits for layer 1 (H=1, D=40)
__global__ void lr1_kernel(const float* __restrict__ feat,
                           const float* __restrict__ al, const float* __restrict__ ar,
                           float* __restrict__ el, float* __restrict__ er, int n)
{
    int nid = blockIdx.x * blockDim.x + threadIdx.x;
    if (nid >= n) return;
    const float* f = feat + (size_t)nid * GAT_C;
    float sl = 0.f, sr = 0.f;
    #pragma unroll
    for (int c = 0; c < GAT_C; ++c) { float v = f[c]; sl += v * al[c]; sr += v * ar[c]; }
    el[nid] = sl; er[nid] = sr;
}

// pass 1: e = leaky_relu(el[src]+er[dst]); segment-max via ordered-int atomicMax
template <int H>
__global__ void edge_attn_max_kernel(const int* __restrict__ src, const int* __restrict__ dst,
                                     const float* __restrict__ el, const float* __restrict__ er,
                                     float* __restrict__ ebuf, int* __restrict__ emax,
                                     int E_)
{
    int e = blockIdx.x * blockDim.x + threadIdx.x;
    if (e >= E_) return;
    int s = src[e], d = dst[e];
    float v[H];
    if constexpr (H == 4) {
        float4 l4 = ((const float4*)el)[s];
        float4 r4 = ((const float4*)er)[d];
        v[0] = l4.x + r4.x; v[1] = l4.y + r4.y;
        v[2] = l4.z + r4.z; v[3] = l4.w + r4.w;
    } else {
        v[0] = el[s] + er[d];
    }
    #pragma unroll
    for (int h = 0; h < H; ++h) {
        float w = v[h] >= 0.f ? v[h] : SLOPE * v[h];
        ebuf[(size_t)h * E_ + e] = w;
        atomicMax(&emax[d * H + h], f2oi(w));
    }
}

// pass 2: a = exp(e - emax[dst]); denom[dst] += a
template <int H>
__global__ void edge_exp_sum_kernel(const int* __restrict__ dst,
                                    float* __restrict__ ebuf, const int* __restrict__ emax,
                                    float* __restrict__ denom, int E_)
{
    int e = blockIdx.x * blockDim.x + threadIdx.x;
    if (e >= E_) return;
    int d = dst[e];
    int m[H];
    if constexpr (H == 4) {
        int4 m4 = ((const int4*)emax)[d];
        m[0] = m4.x; m[1] = m4.y; m[2] = m4.z; m[3] = m4.w;
    } else {
        m[0] = emax[d];
    }
    #pragma unroll
    for (int h = 0; h < H; ++h) {
        float a = __expf(ebuf[(size_t)h * E_ + e] - oi2f(m[h]));
        ebuf[(size_t)h * E_ + e] = a;
        atomicAdd(&denom[d * H + h], a);
    }
}

// pass 3 (layer 0): warp per edge, lane = feature dim within head
__global__ __launch_bounds__(256) void edge_aggr0_kernel(
    const int* __restrict__ src, const int* __restrict__ dst,
    const float* __restrict__ feat, const float* __restrict__ ebuf,
    const float* __restrict__ denom, float* __restrict__ out, int E_)
{
    int warp = (blockIdx.x * blockDim.x + threadIdx.x) >> 5;
    int lane = threadIdx.x & 31;
    if (warp >= E_) return;
    int s = __builtin_amdgcn_readfirstlane(src[warp]);
    int d = __builtin_amdgcn_readfirstlane(dst[warp]);
    __builtin_prefetch(&feat[(size_t)s * GAT_HID + lane], 0, 0);  // global_prefetch_b8
    #pragma unroll
    for (int h = 0; h < GAT_H0; ++h) {
        float a     = ebuf[(size_t)h * E_ + warp];
        float dn    = denom[d * GAT_H0 + h];
        float alpha = a / fmaxf(dn, 1e-9f);
        float v = feat[(size_t)s * GAT_HID + h * GAT_D0 + lane] * alpha;
        atomicAdd(&out[(size_t)d * GAT_HID + h * GAT_D0 + lane], v);
    }
}

// pass 3 (layer 1): warp per edge, D=40 (lanes cover c and c+32)
__global__ __launch_bounds__(256) void edge_aggr1_kernel(
    const int* __restrict__ src, const int* __restrict__ dst,
    const float* __restrict__ feat, const float* __restrict__ ebuf,
    const float* __restrict__ denom, float* __restrict__ out, int E_)
{
    int warp = (blockIdx.x * blockDim.x + threadIdx.x) >> 5;
    int lane = threadIdx.x & 31;
    if (warp >= E_) return;
    int s = __builtin_amdgcn_readfirstlane(src[warp]);
    int d = __builtin_amdgcn_readfirstlane(dst[warp]);
    float alpha = ebuf[warp] / fmaxf(denom[d], 1e-9f);
    for (int c = lane; c < GAT_C; c += 32) {
        atomicAdd(&out[(size_t)d * GAT_C + c], feat[(size_t)s * GAT_C + c] * alpha);
    }
}

// ELU in place: h = elu(agg0)
__global__ void elu_kernel(float* __restrict__ p, size_t count) {
    size_t i = (size_t)blockIdx.x * blockDim.x + threadIdx.x;
    if (i >= count) return;
    float v = p[i];
    p[i] = v > 0.f ? v : (__expf(v) - 1.f);
}

extern "C" void kernel_launch(void* const* d_in, const int* in_sizes, int n_in,
                              void* d_out, int out_size, void* d_ws, size_t ws_size,
                              hipStream_t stream) {
    const float* x   = (const float*)d_in[0];
    const int*   src = (const int*)d_in[1];
    const int*   dst = (const int*)d_in[2];
    const float* W0  = (const float*)d_in[3];
    const float* al0 = (const float*)d_in[4];
    const float* ar0 = (const float*)d_in[5];
    const float* W1  = (const float*)d_in[6];
    const float* al1 = (const float*)d_in[7];
    const float* ar1 = (const float*)d_in[8];
    float* out = (float*)d_out;

    const int n  = in_sizes[0] / GAT_IN;   // N nodes
    const int E_ = in_sizes[1];            // edges

    // ---- workspace layout (256B aligned) ----
    char* ws = (char*)d_ws;
    size_t off = 0;
    auto alloc = [&](size_t bytes) -> char* {
        char* p = ws + off;
        off = (off + bytes + 255) & ~(size_t)255;
        return p;
    };
    float* feat0  = (float*)alloc((size_t)n * GAT_HID * 4);
    float* agg0   = (float*)alloc((size_t)n * GAT_HID * 4);   // becomes h after ELU
    float* feat1  = (float*)alloc((size_t)n * GAT_C * 4);
    float* ebuf0  = (float*)alloc((size_t)E_ * GAT_H0 * 4);
    float* ebuf1  = (float*)alloc((size_t)E_ * 4);
    float* el0    = (float*)alloc((size_t)n * GAT_H0 * 4);
    float* er0    = (float*)alloc((size_t)n * GAT_H0 * 4);
    int*   emax0  = (int*)  alloc((size_t)n * GAT_H0 * 4);
    float* denom0 = (float*)alloc((size_t)n * GAT_H0 * 4);
    float* el1    = (float*)alloc((size_t)n * 4);
    float* er1    = (float*)alloc((size_t)n * 4);
    int*   emax1  = (int*)  alloc((size_t)n * 4);
    float* denom1 = (float*)alloc((size_t)n * 4);
    (void)ws_size; (void)n_in; (void)out_size;

    // ---- init accumulators (every call: harness does not re-poison) ----
    (void)hipMemsetAsync(denom0, 0, (size_t)n * GAT_H0 * 4, stream);
    (void)hipMemsetAsync(denom1, 0, (size_t)n * 4, stream);
    (void)hipMemsetAsync(agg0,   0, (size_t)n * GAT_HID * 4, stream);
    (void)hipMemsetAsync(out,    0, (size_t)n * GAT_C * 4, stream);
    fill_i32_kernel<<<(n * GAT_H0 + 255) / 256, 256, 0, stream>>>(emax0, (int)0x80000000, n * GAT_H0);
    fill_i32_kernel<<<(n + 255) / 256, 256, 0, stream>>>(emax1, (int)0x80000000, n);

    // ---- layer 0 ----
    gemm0_kernel<<<(n + 15) / 16, 256, 0, stream>>>(x, W0, feat0, n);
    lr0_kernel<<<(n * GAT_H0 + 255) / 256, 256, 0, stream>>>(feat0, al0, ar0, el0, er0, n);
    edge_attn_max_kernel<GAT_H0><<<(E_ + 255) / 256, 256, 0, stream>>>(src, dst, el0, er0, ebuf0, emax0, E_);
    edge_exp_sum_kernel<GAT_H0><<<(E_ + 255) / 256, 256, 0, stream>>>(dst, ebuf0, emax0, denom0, E_);
    edge_aggr0_kernel<<<(E_ + 7) / 8, 256, 0, stream>>>(src, dst, feat0, ebuf0, denom0, agg0, E_);
    elu_kernel<<<(int)(((size_t)n * GAT_HID + 255) / 256), 256, 0, stream>>>(agg0, (size_t)n * GAT_HID);

    // ---- layer 1 ----
    gemm1_kernel<<<(n + 15) / 16, 256, 0, stream>>>(agg0, W1, feat1, n);
    lr1_kernel<<<(n + 255) / 256, 256, 0, stream>>>(feat1, al1, ar1, el1, er1, n);
    edge_attn_max_kernel<1><<<(E_ + 255) / 256, 256, 0, stream>>>(src, dst, el1, er1, ebuf1, emax1, E_);
    edge_exp_sum_kernel<1><<<(E_ + 255) / 256, 256, 0, stream>>>(dst, ebuf1, emax1, denom1, E_);
    edge_aggr1_kernel<<<(E_ + 7) / 8, 256, 0, stream>>>(src, dst, feat1, ebuf1, denom1, out, E_);
    // mean over H1=1 heads is the identity, so `out` is final [N, 40].
}